// DeformConv_45878840655977
// MI455X (gfx1250) — compile-verified
//
#include <hip/hip_runtime.h>
#include <hip/hip_bf16.h>

// Deformable-conv (DCNv2) modulated bilinear sampling on gfx1250.
// One wave32 per (output pixel, group). Reduction over the 4 bilinear
// corners is done with V_WMMA_F32_16X16X4_F32 (K=4 == 4 corners),
// 16 channels per WMMA, 4 channel-tiles x 9 taps = 36 WMMAs per wave.
// Tap geometry is computed lane-parallel (lanes 0..8) and broadcast once
// via v_readlane (no LDS traffic), hoisted out of the channel-tile loop.

typedef __attribute__((ext_vector_type(2))) float v2f;
typedef __attribute__((ext_vector_type(8))) float v8f;

#define Bn   16
#define Hn   56
#define Wn   56
#define Gn   4
#define CGn  64
#define Kn   9
#define Cn   (Gn * CGn)      // 256 channels
#define OMC  (Gn * Kn * 3)   // 108 offset/mask floats per pixel

__device__ __forceinline__ int bcasti(int v, int l) {
    return __builtin_amdgcn_readlane(v, l);
}
__device__ __forceinline__ float bcastf(float v, int l) {
    return __int_as_float(__builtin_amdgcn_readlane(__float_as_int(v), l));
}

__global__ __launch_bounds__(256)
void dcn_sample_wmma(const float* __restrict__ x,
                     const float* __restrict__ om,
                     float* __restrict__ out)
{
    const int lane = threadIdx.x & 31;
    const int wave = threadIdx.x >> 5;

    // pixel-group index: 16*56*56*4 = 200704 == gridDim.x(25088) * 8 waves
    const int i = blockIdx.x * 8 + wave;
    const int g = i & (Gn - 1);
    int pix = i >> 2;
    const int w = pix % Wn;  pix /= Wn;
    const int h = pix % Hn;
    const int b = pix / Hn;

    // ---- per-tap geometry: lanes 0..8 own tap t = lane (others compute t=0, unused)
    const int t = (lane < Kn) ? lane : 0;
    const int ombase = ((b * Hn + h) * Wn + w) * OMC + g * (Kn * 3);

    // prefetch neighboring pixel's offset/mask block (pure hint)
    __builtin_prefetch(om + ombase + OMC, 0, 1);

    const float offx = om[ombase + 2 * t];
    const float offy = om[ombase + 2 * t + 1];
    const float msk  = om[ombase + 2 * Kn + t];

    const float gy = (float)(t / 3 - 1);
    const float gx = (float)(t % 3 - 1);

    // sample position in PADDED coords (PAD=1, STRIDE=1, DIL=1):
    // sy = (1 + 0.5 + h) + gy + offy - 0.5
    const float sy = (float)(h + 1) + gy + offy;
    const float sx = (float)(w + 1) + gx + offx;

    const float y0f = floorf(sy), x0f = floorf(sx);
    const float wy1 = sy - y0f,   wx1 = sx - x0f;
    const float wy0 = 1.0f - wy1, wx0 = 1.0f - wx1;
    const int y0 = (int)y0f, x0 = (int)x0f;
    const int y1 = y0 + 1,   x1 = x0 + 1;

    // padded ring rows/cols are zero, so only padded coords 1..56 contribute
    const bool vy0 = (y0 >= 1) && (y0 <= Hn);
    const bool vy1 = (y1 >= 1) && (y1 <= Hn);
    const bool vx0 = (x0 >= 1) && (x0 <= Wn);
    const bool vx1 = (x1 >= 1) && (x1 <= Wn);

    const float w00 = (vy0 && vx0) ? wy0 * wx0 * msk : 0.0f;  // K=0 (y0,x0)
    const float w01 = (vy0 && vx1) ? wy0 * wx1 * msk : 0.0f;  // K=1 (y0,x1)
    const float w10 = (vy1 && vx0) ? wy1 * wx0 * msk : 0.0f;  // K=2 (y1,x0)
    const float w11 = (vy1 && vx1) ? wy1 * wx1 * msk : 0.0f;  // K=3 (y1,x1)

    // unpadded clamped coords (weight already 0 when out of bounds)
    const int yu0 = min(max(y0 - 1, 0), Hn - 1);
    const int yu1 = min(max(y1 - 1, 0), Hn - 1);
    const int xu0 = min(max(x0 - 1, 0), Wn - 1);
    const int xu1 = min(max(x1 - 1, 0), Wn - 1);

    // flat element offsets of channel 0 of this group at each corner
    const int a00 = ((b * Hn + yu0) * Wn + xu0) * Cn + g * CGn;
    const int a01 = ((b * Hn + yu0) * Wn + xu1) * Cn + g * CGn;
    const int a10 = ((b * Hn + yu1) * Wn + xu0) * Cn + g * CGn;
    const int a11 = ((b * Hn + yu1) * Wn + xu1) * Cn + g * CGn;

    const int chlo = lane & 15;          // channel within 16-wide tile
    const bool lowhalf = (lane < 16);

    // ---- broadcast tap data once (v_readlane -> SGPRs), build per-lane
    //      load offsets and B-matrix registers; tap-indexed arrays are
    //      fully unrolled -> live in registers.
    int off0[Kn], off1[Kn];
    v2f bmat[Kn];
    #pragma unroll
    for (int tap = 0; tap < Kn; ++tap) {
        const int   A00 = bcasti(a00, tap);
        const int   A01 = bcasti(a01, tap);
        const int   A10 = bcasti(a10, tap);
        const int   A11 = bcasti(a11, tap);
        const float W00 = bcastf(w00, tap);
        const float W01 = bcastf(w01, tap);
        const float W10 = bcastf(w10, tap);
        const float W11 = bcastf(w11, tap);

        // A matrix 16x4 f32: lanes 0-15 -> VGPR0=K0 (y0,x0); VGPR1=K1 (y0,x1)
        //                    lanes16-31 -> VGPR0=K2 (y1,x0); VGPR1=K3 (y1,x1)
        off0[tap] = (lowhalf ? A00 : A10) + chlo;
        off1[tap] = (lowhalf ? A01 : A11) + chlo;

        // B matrix 4x16 f32, only column n=0 nonzero:
        // VGPR0: lane0=K0 weight, lane16=K2; VGPR1: lane0=K1, lane16=K3
        v2f bb;
        bb.x = (lane == 0) ? W00 : ((lane == 16) ? W10 : 0.0f);
        bb.y = (lane == 0) ? W01 : ((lane == 16) ? W11 : 0.0f);
        bmat[tap] = bb;
    }

    const int outpix = ((b * Hn + h) * Wn + w) * Cn + g * CGn;

    #pragma unroll
    for (int mt = 0; mt < 4; ++mt) {     // 4 channel tiles of 16 (CG=64)
        v8f acc = {0.f, 0.f, 0.f, 0.f, 0.f, 0.f, 0.f, 0.f};

        #pragma unroll
        for (int tap = 0; tap < Kn; ++tap) {
            // mt*16 elements = mt*64 bytes folds into the load's imm offset
            v2f a;
            a.x = x[off0[tap] + mt * 16];
            a.y = x[off1[tap] + mt * 16];

            acc = __builtin_amdgcn_wmma_f32_16x16x4_f32(
                /*neg_a=*/false, a, /*neg_b=*/false, bmat[tap],
                /*c_mod=*/(short)0, acc, /*reuse_a=*/false, /*reuse_b=*/false);
        }

        // D column n=0: lane 0 VGPR r -> M=r (ch mt*16+0..7),
        //               lane16 VGPR r -> M=r+8 (ch mt*16+8..15)
        if (lane == 0 || lane == 16) {
            float* p = out + outpix + mt * 16 + (lowhalf ? 0 : 8);
            float4 lo = make_float4(acc[0], acc[1], acc[2], acc[3]);
            float4 hi = make_float4(acc[4], acc[5], acc[6], acc[7]);
            *(float4*)(p)     = lo;   // 16B-aligned b128 stores
            *(float4*)(p + 4) = hi;
        }
    }
}

extern "C" void kernel_launch(void* const* d_in, const int* in_sizes, int n_in,
                              void* d_out, int out_size, void* d_ws, size_t ws_size,
                              hipStream_t stream) {
    const float* x  = (const float*)d_in[0];
    const float* om = (const float*)d_in[1];
    float* out = (float*)d_out;

    // 16*56*56*4 pixel-groups, 8 waves (256 threads) per block
    const int total_pg = Bn * Hn * Wn * Gn;      // 200704
    const int blocks = total_pg / 8;             // 25088, exact
    dcn_sample_wmma<<<blocks, 256, 0, stream>>>(x, om, out);
}